// UnrestrictedDet_32469952758281
// MI455X (gfx1250) — compile-verified
//
#include <hip/hip_runtime.h>
#include <math.h>

// ---------------------------------------------------------------------------
// Batched slogdet of gathered 128x128 matrices (UnrestrictedDet).
// One workgroup (256 threads / 8 waves, wave32) per batch sample; it runs
// blocked LU with partial pivoting on the "up" matrix then the "dn" matrix
// in the same LDS buffer. Trailing updates use V_WMMA_F32_16X16X4_F32.
// ---------------------------------------------------------------------------

typedef __attribute__((ext_vector_type(2))) float v2f;
typedef __attribute__((ext_vector_type(8))) float v8f;

#define N        128      // matrix dimension (NUP == NDN == 128)
#define SROW     132      // padded LDS row stride in floats (528 B, 16B aligned)
#define NB       16       // LU panel width == WMMA tile edge
#define NTHREADS 256
#define NWAVES   8

struct LuScalars {
    unsigned long long pivkey;  // (abs-bits << 32) | row
    float sign;
    float logabs;
};

// In-place blocked LU with partial pivoting on sA (N x N, row stride SROW).
// Accumulates determinant sign and log|det| into sc. All barriers internal.
__device__ __forceinline__ void lu_slogdet_inplace(float* sA, LuScalars* sc, int tid)
{
    const int lane   = tid & 31;
    const int wave   = tid >> 5;
    const int rowsel = lane >> 4;   // 0 for lanes 0-15, 1 for lanes 16-31
    const int lan16  = lane & 15;

    for (int p = 0; p < N / NB; ++p) {
        const int c0   = p * NB;
        const int cend = c0 + NB;

        // ---------- panel factorization (columns c0..cend-1) ----------
        for (int j = c0; j < cend; ++j) {
            if (tid == 0) sc->pivkey = 0ull;
            __syncthreads();

            // pivot search over rows j..N-1 of column j (ds_max_u64)
            {
                int r = j + tid;
                if (r < N) {
                    float v = sA[r * SROW + j];
                    unsigned long long key =
                        ((unsigned long long)__float_as_uint(fabsf(v)) << 32) |
                        (unsigned int)r;
                    atomicMax(&sc->pivkey, key);
                }
            }
            __syncthreads();

            unsigned long long k = sc->pivkey;
            int   piv  = (int)(k & 0xffffffffull);
            float pval = sA[piv * SROW + j];   // signed pivot, read pre-swap
            __syncthreads();                    // all reads done before swap writes

            // swap full rows j <-> piv (thread per column)
            if (piv != j && tid < N) {
                float a = sA[j * SROW + tid];
                float b = sA[piv * SROW + tid];
                sA[j * SROW + tid]   = b;
                sA[piv * SROW + tid] = a;
            }
            if (tid == 0) {
                float s = sc->sign;
                if (piv != j)    s = -s;       // row-swap parity
                if (pval < 0.0f) s = -s;       // pivot sign
                sc->sign    = s;
                sc->logabs += logf(fabsf(pval));
            }
            __syncthreads();

            // multipliers: column j below diagonal
            float inv = 1.0f / pval;
            {
                int r2 = j + 1 + tid;
                if (r2 < N) sA[r2 * SROW + j] *= inv;
            }
            __syncthreads();

            // rank-1 update restricted to remaining PANEL columns
            int ncols = cend - (j + 1);
            if (ncols > 0) {
                int nrows = N - (j + 1);
                for (int i = tid; i < nrows * ncols; i += NTHREADS) {
                    int rr = j + 1 + i / ncols;
                    int cc = j + 1 + i % ncols;
                    sA[rr * SROW + cc] -= sA[rr * SROW + j] * sA[j * SROW + cc];
                }
            }
            __syncthreads();
        }

        const int ntc = N - cend;           // trailing width
        if (ntc == 0) break;                // p == 7: done

        // ---------- U12 = L11^{-1} * A12 (unit lower triangular solve) ----------
        // One thread per trailing column, fully register-resident.
        if (tid < ntc) {
            int c = cend + tid;
            float a[NB];
            #pragma unroll
            for (int i = 0; i < NB; ++i) a[i] = sA[(c0 + i) * SROW + c];
            #pragma unroll
            for (int i = 1; i < NB; ++i) {
                float s = a[i];
                #pragma unroll
                for (int kk = 0; kk < NB; ++kk)   // only kk < i contribute
                    if (kk < i)
                        s -= sA[(c0 + i) * SROW + (c0 + kk)] * a[kk];
                a[i] = s;
            }
            #pragma unroll
            for (int i = 0; i < NB; ++i) sA[(c0 + i) * SROW + c] = a[i];
        }
        __syncthreads();

        // ---------- trailing update: A22 -= L21 * U12 via WMMA f32 16x16x4 ----------
        // Wave-uniform tile loop => EXEC is all ones at every WMMA.
        const int ntil = ntc >> 4;
        for (int t = wave; t < ntil * ntil; t += NWAVES) {
            int tr = t / ntil;
            int tc = t % ntil;
            int rb = cend + 16 * tr;
            int cb = cend + 16 * tc;

            // C/D 16x16 f32 layout: VGPR v, lanes 0-15 -> M=v, lanes 16-31 -> M=v+8
            int lrow = rb + 8 * rowsel;
            int lcol = cb + lan16;
            v8f c;
            #pragma unroll
            for (int v = 0; v < 8; ++v) c[v] = sA[(lrow + v) * SROW + lcol];

            #pragma unroll
            for (int kk = 0; kk < 4; ++kk) {
                int k0 = c0 + 4 * kk;
                int ak = k0 + 2 * rowsel;   // K pair selected by lane half
                v2f a, bm;
                // A 16x4 layout: lane -> M=lane&15; VGPR0/1 -> K = 2*(lane>=16)+{0,1}
                // negate L21 so the MAC performs C - L*U (f32 WMMA has no A-neg bit)
                a[0] = -sA[(rb + lan16) * SROW + ak];
                a[1] = -sA[(rb + lan16) * SROW + ak + 1];
                // B 4x16 layout: VGPR v, lanes 0-15 -> K=v, lanes 16-31 -> K=v+2
                bm[0] = sA[ak * SROW + lcol];
                bm[1] = sA[(ak + 1) * SROW + lcol];
                c = __builtin_amdgcn_wmma_f32_16x16x4_f32(
                        /*neg_a=*/false, a, /*neg_b=*/false, bm,
                        /*c_mod=*/(short)0, c, /*reuse_a=*/false, /*reuse_b=*/false);
            }

            #pragma unroll
            for (int v = 0; v < 8; ++v) sA[(lrow + v) * SROW + lcol] = c[v];
        }
        __syncthreads();
    }
}

__global__ __launch_bounds__(NTHREADS)
void unrestricted_det_kernel(const float* __restrict__ Uup,
                             const float* __restrict__ Udn,
                             const int*   __restrict__ idx_up,
                             const int*   __restrict__ idx_dn,
                             float* __restrict__ out, int B)
{
    extern __shared__ float sA[];        // N x SROW floats (dynamic: 67,584 B)
    __shared__ LuScalars sc;

    const int b   = blockIdx.x;
    const int tid = threadIdx.x;

    float accSign = 1.0f;
    float accLog  = 0.0f;

    for (int spin = 0; spin < 2; ++spin) {
        const float* U   = spin ? Udn : Uup;
        const int*   idx = (spin ? idx_dn : idx_up) + (size_t)b * N;

        // gather occupied rows into LDS (coalesced float4 loads, b128 LDS stores)
        const float4* U4 = reinterpret_cast<const float4*>(U);
        for (int i = tid; i < N * (N / 4); i += NTHREADS) {
            int row = i >> 5;            // N/4 == 32 float4 per row
            int c4  = i & 31;
            float4 v = U4[(size_t)idx[row] * (N / 4) + c4];
            *reinterpret_cast<float4*>(&sA[row * SROW + c4 * 4]) = v;
        }
        if (tid == 0) { sc.sign = 1.0f; sc.logabs = 0.0f; }
        __syncthreads();

        lu_slogdet_inplace(sA, &sc, tid);
        // last barrier inside lu_slogdet_inplace makes sc visible

        if (tid == 0) { accSign *= sc.sign; accLog += sc.logabs; }
        __syncthreads();                 // before reusing sA for next spin
    }

    if (tid == 0) {
        out[b]     = accSign;            // sign_up * sign_dn
        out[B + b] = accLog;             // logabs_up + logabs_dn
    }
}

extern "C" void kernel_launch(void* const* d_in, const int* in_sizes, int n_in,
                              void* d_out, int out_size, void* d_ws, size_t ws_size,
                              hipStream_t stream)
{
    const float* Uup    = (const float*)d_in[0];   // [256,128]
    const float* Udn    = (const float*)d_in[1];   // [256,128]
    const int*   idx_up = (const int*)d_in[2];     // [B,128]
    const int*   idx_dn = (const int*)d_in[3];     // [B,128]
    float*       out    = (float*)d_out;           // [sign(B), logabs(B)]

    const int B = in_sizes[2] / N;

    const size_t shmem = (size_t)N * SROW * sizeof(float);
    // dynamic-LDS opt-in above default cap; idempotent + deterministic
    (void)hipFuncSetAttribute((const void*)unrestricted_det_kernel,
                              hipFuncAttributeMaxDynamicSharedMemorySize,
                              (int)shmem);

    unrestricted_det_kernel<<<dim3(B), dim3(NTHREADS), shmem, stream>>>(
        Uup, Udn, idx_up, idx_dn, out, B);
}